// AdaPool2D_79482664780428
// MI455X (gfx1250) — compile-verified
//
#include <hip/hip_runtime.h>
#include <stdint.h>

// AdaPool2D for input [32,224,224,64] f32, 2x2/stride2 -> [32,112,112,64] f32.
// Memory-bound-ish (~514MB traffic, ~22us floor @23.3TB/s) and VALU-balanced.
// Uses gfx1250 async tensor-class data movement: global_load_async_to_lds_b128
// double-buffered per tile, tracked with ASYNCcnt (s_wait_asynccnt).

typedef float v4f __attribute__((ext_vector_type(4)));  // native clang vector

#define B_   32
#define WI_  224
#define HI_  224
#define C_   64
#define WO_  112
#define HO_  112
#define TQ_  16              // output pixels per tile (along inner spatial dim)
#define NT_  (HO_ / TQ_)     // 7 tiles per output row
// staged tile: 2 input rows x 32 input pixels x 64 ch x 4B = 16KB per buffer

__device__ __forceinline__ float adapool_one(float a, float b, float c, float d,
                                             float m, float om) {
    // exponential-maximum pool: softmax(v) . v
    float mx = fmaxf(fmaxf(a, b), fmaxf(c, d));
    float ea = __expf(a - mx), eb = __expf(b - mx);
    float ec = __expf(c - mx), ed = __expf(d - mx);
    float rs = __builtin_amdgcn_rcpf(ea + eb + ec + ed);
    float em = (a * ea + b * eb + c * ec + d * ed) * rs;

    // eDSCW pool: dsc_k = 2*avg*v_k / (avg^2 + v_k^2), w = softmax(dsc), w . v
    float avg = 0.25f * ((a + b) + (c + d));
    float a2  = avg * avg;
    float t2  = 2.0f * avg;
    float da = (t2 * a) * __builtin_amdgcn_rcpf(fmaf(a, a, a2));
    float db = (t2 * b) * __builtin_amdgcn_rcpf(fmaf(b, b, a2));
    float dc = (t2 * c) * __builtin_amdgcn_rcpf(fmaf(c, c, a2));
    float dd = (t2 * d) * __builtin_amdgcn_rcpf(fmaf(d, d, a2));
    float dmx = fmaxf(fmaxf(da, db), fmaxf(dc, dd));
    float wa = __expf(da - dmx), wb = __expf(db - dmx);
    float wc = __expf(dc - dmx), wd = __expf(dd - dmx);
    float rw = __builtin_amdgcn_rcpf(wa + wb + wc + wd);
    float dp = (a * wa + b * wb + c * wc + d * wd) * rw;

    return em * m + dp * om;
}

__global__ void __launch_bounds__(256)
adapool2d_async_kernel(const float* __restrict__ x,
                       const float* __restrict__ mask,
                       float* __restrict__ out) {
    // [buf][input row in pair][input pixel 0..31][channel 0..63]
    __shared__ __align__(16) float smem[2][2][32][C_];

    const int t   = threadIdx.x;             // 0..255
    const int row = blockIdx.x;              // 0 .. B*WO-1
    const int b   = row / WO_;
    const int ow  = row - b * WO_;           // output index along first spatial dim

    const float* r0 = x + ((size_t)(b * WI_ + 2 * ow)) * (size_t)(HI_ * C_);
    const float* r1 = r0 + (size_t)(HI_ * C_);

    // Low 32 bits of a generic pointer to LDS == byte offset within wave LDS
    // (CDNA5 aperture: LDS address is the pointer's low 32 bits).
    const uint32_t lds_base = (uint32_t)(uintptr_t)(&smem[0][0][0][0]);

    const float m  = mask[0];
    const float om = 1.0f - m;

    // Stage one 16KB tile (2 x 8KB contiguous row segments) into LDS buffer
    // `buf` with 4 async 16B copies per thread (1024 chunks total).
    auto issue_tile = [&](int q, int buf) {
        const int hbase = q * 32 * C_;       // float offset of tile within row
#pragma unroll
        for (int i = 0; i < 4; ++i) {
            const int k = t + i * 256;                   // chunk id 0..1023
            const float* g = ((k < 512) ? r0 : r1) + hbase + (k & 511) * 4;
            const uint32_t l = lds_base + (uint32_t)buf * 16384u + (uint32_t)k * 16u;
            asm volatile("global_load_async_to_lds_b128 %0, %1, off"
                         :: "v"(l), "v"(g) : "memory");
        }
    };

    issue_tile(0, 0);

    const int p  = t >> 4;        // output pixel within tile: 0..15
    const int cg = t & 15;        // channel group (4 channels): 0..15
    float* orow = out + ((size_t)(b * WO_ + ow)) * (size_t)(HO_ * C_);

#pragma unroll 1
    for (int q = 0; q < NT_; ++q) {
        const int buf = q & 1;
        if (q + 1 < NT_) {
            issue_tile(q + 1, buf ^ 1);          // prefetch next tile
            // in-order ASYNCcnt: <=4 outstanding => current tile's copies done
            asm volatile("s_wait_asynccnt 4" ::: "memory");
        } else {
            asm volatile("s_wait_asynccnt 0" ::: "memory");
        }
        __syncthreads();                          // all waves' chunks visible

        const v4f va = *(const v4f*)&smem[buf][0][2 * p    ][cg * 4];
        const v4f vb = *(const v4f*)&smem[buf][0][2 * p + 1][cg * 4];
        const v4f vc = *(const v4f*)&smem[buf][1][2 * p    ][cg * 4];
        const v4f vd = *(const v4f*)&smem[buf][1][2 * p + 1][cg * 4];

        v4f r;
        r.x = adapool_one(va.x, vb.x, vc.x, vd.x, m, om);
        r.y = adapool_one(va.y, vb.y, vc.y, vd.y, m, om);
        r.z = adapool_one(va.z, vb.z, vc.z, vd.z, m, om);
        r.w = adapool_one(va.w, vb.w, vc.w, vd.w, m, om);

        // streaming output: non-temporal 128-bit store, fully coalesced
        __builtin_nontemporal_store(
            r, (v4f*)(orow + (size_t)(q * TQ_ + p) * C_ + cg * 4));

        __syncthreads();   // buffer buf may be overwritten next iteration
    }
}

extern "C" void kernel_launch(void* const* d_in, const int* in_sizes, int n_in,
                              void* d_out, int out_size, void* d_ws, size_t ws_size,
                              hipStream_t stream) {
    (void)in_sizes; (void)n_in; (void)d_ws; (void)ws_size; (void)out_size;
    const float* x    = (const float*)d_in[0];
    const float* mask = (const float*)d_in[1];
    float*       out  = (float*)d_out;

    dim3 grid(B_ * WO_);   // one block per output row: 3584 blocks
    dim3 block(256);       // 8 waves (wave32)
    adapool2d_async_kernel<<<grid, block, 0, stream>>>(x, mask, out);
}